// TopK_custom_18992345383426
// MI455X (gfx1250) — compile-verified
//
#include <hip/hip_runtime.h>
#include <stdint.h>

#define BS   256
#define NN   8192
#define KK   16
#define KP1  17
#define EPSI 0.1f
#define ITERS 200
#define ROWS 8          // NN / 1024 rows per thread
#define RSTR 20         // padded stride for wave partials

typedef float        v2f   __attribute__((ext_vector_type(2)));
typedef float        v8f   __attribute__((ext_vector_type(8)));
typedef float        f4    __attribute__((ext_vector_type(4)));
typedef unsigned int u32x4 __attribute__((ext_vector_type(4)));
typedef int          i32x4 __attribute__((ext_vector_type(4)));
typedef int          i32x8 __attribute__((ext_vector_type(8)));

// ---- wave32 cross-lane helpers: DPP ROW_XMASK (VALU) instead of ds_bpermute ----
template <int CTRL, int MASK>
__device__ __forceinline__ float dppx(float x) {
#if defined(__HIP_DEVICE_COMPILE__) && __has_builtin(__builtin_amdgcn_update_dpp)
    return __int_as_float(__builtin_amdgcn_update_dpp(
        0, __float_as_int(x), CTRL, 0xf, 0xf, true));
#else
    return __shfl_xor(x, MASK, 32);
#endif
}
__device__ __forceinline__ float xor16v(float x) {
#if defined(__HIP_DEVICE_COMPILE__) && __has_builtin(__builtin_amdgcn_permlanex16)
    int xi = __float_as_int(x);
    return __int_as_float(__builtin_amdgcn_permlanex16(
        xi, xi, 0x76543210, 0xfedcba98, false, false));
#else
    return __shfl_xor(x, 16, 32);
#endif
}
__device__ __forceinline__ float wave_sum(float x) {
    x += dppx<0x161, 1>(x);   // row_xmask 1
    x += dppx<0x162, 2>(x);   // row_xmask 2
    x += dppx<0x164, 4>(x);   // row_xmask 4
    x += dppx<0x168, 8>(x);   // row_xmask 8
    x += xor16v(x);           // cross-half
    return x;
}
__device__ __forceinline__ float wave_max(float x) {
    x = fmaxf(x, dppx<0x161, 1>(x));
    x = fmaxf(x, dppx<0x162, 2>(x));
    x = fmaxf(x, dppx<0x164, 4>(x));
    x = fmaxf(x, dppx<0x168, 8>(x));
    x = fmaxf(x, xor16v(x));
    return x;
}
__device__ __forceinline__ float wave_min(float x) {
    x = fminf(x, dppx<0x161, 1>(x));
    x = fminf(x, dppx<0x162, 2>(x));
    x = fminf(x, dppx<0x164, 4>(x));
    x = fminf(x, dppx<0x168, 8>(x));
    x = fminf(x, xor16v(x));
    return x;
}
__device__ __forceinline__ float bcast_lane(float x, int lane) {
#if defined(__HIP_DEVICE_COMPILE__) && __has_builtin(__builtin_amdgcn_readlane)
    return __int_as_float(__builtin_amdgcn_readlane(__float_as_int(x), lane));
#else
    return __shfl(x, lane, 32);
#endif
}

// ---------------------------------------------------------------------------
// Kernel 1a: per-block partial min/max/neg-inf-flag over all scores
// ---------------------------------------------------------------------------
__global__ __launch_bounds__(256) void minmax_part(const float* __restrict__ s,
                                                   float* __restrict__ ws, int ntot) {
    int tid = blockIdx.x * blockDim.x + threadIdx.x;
    float mx = -3.4e38f, mn = 3.4e38f, ng = 0.0f;
    for (int i = tid; i < ntot; i += gridDim.x * blockDim.x) {
        float v = s[i];
        bool neg = (v == -__builtin_inff());
        mx = fmaxf(mx, v);                       // raw max (matches jnp.max(s_d))
        mn = fminf(mn, neg ? 3.4e38f : v);       // min over non -inf
        ng = neg ? 1.0f : ng;
    }
    mx = wave_max(mx); mn = wave_min(mn); ng = wave_max(ng);
    __shared__ float smx[8], smn[8], sng[8];
    int lane = threadIdx.x & 31, wid = threadIdx.x >> 5;
    if (lane == 0) { smx[wid] = mx; smn[wid] = mn; sng[wid] = ng; }
    __syncthreads();
    if (threadIdx.x == 0) {
        #pragma unroll
        for (int w = 1; w < 8; ++w) {
            smx[0] = fmaxf(smx[0], smx[w]);
            smn[0] = fminf(smn[0], smn[w]);
            sng[0] = fmaxf(sng[0], sng[w]);
        }
        ws[16 +       blockIdx.x] = smx[0];
        ws[16 + 256 + blockIdx.x] = smn[0];
        ws[16 + 512 + blockIdx.x] = sng[0];
    }
}

// ---------------------------------------------------------------------------
// Kernel 1b: fold 256 partials -> ws[0]=max, ws[1]=min(non-inf), ws[2]=has_neg
// ---------------------------------------------------------------------------
__global__ __launch_bounds__(256) void minmax_final(float* __restrict__ ws) {
    int t = threadIdx.x;
    float mx = ws[16 + t];
    float mn = ws[16 + 256 + t];
    float ng = ws[16 + 512 + t];
    mx = wave_max(mx); mn = wave_min(mn); ng = wave_max(ng);
    __shared__ float smx[8], smn[8], sng[8];
    int lane = t & 31, wid = t >> 5;
    if (lane == 0) { smx[wid] = mx; smn[wid] = mn; sng[wid] = ng; }
    __syncthreads();
    if (t == 0) {
        #pragma unroll
        for (int w = 1; w < 8; ++w) {
            smx[0] = fmaxf(smx[0], smx[w]);
            smn[0] = fminf(smn[0], smn[w]);
            sng[0] = fmaxf(sng[0], sng[w]);
        }
        ws[0] = smx[0]; ws[1] = smn[0]; ws[2] = sng[0];
    }
}

// ---------------------------------------------------------------------------
// Kernel 2: one workgroup (32 wave32s) per batch row; full Sinkhorn in-regs.
//   G_ij = A_i * B_j * t_i^(16-j); A and B cancel in the iteration, leaving
//   p_i = sum_j w_j t^(16-j)  (Horner), moments M_m = sum_i (mu/p_i) t^m,
//   w_j <- nu_j / M_(16-j).  Output Gamma*n = t^(16-j) * w_j / p_i.
// Every wave redundantly reduces the 32x17 partial-moment tile with 8 chained
// V_WMMA_F32_16X16X4_F32 (ones-matrix trick) and rebuilds w via v_readlane,
// so only ONE barrier per iteration is needed (ping-pong partial buffers).
// ---------------------------------------------------------------------------
__global__ __launch_bounds__(1024, 1) void sinkhorn_topk(const float* __restrict__ scores,
                                                         float* __restrict__ out,
                                                         const float* __restrict__ ws) {
    const int b    = blockIdx.x;
    const int tid  = threadIdx.x;
    const int lane = tid & 31;
    const int wid  = tid >> 5;

    __shared__ __align__(16) float s_lds[NN];          // 32 KB staged scores
    __shared__ __align__(16) float red[2][32 * RSTR];  // ping-pong moment partials

    // ---- stage this batch's 8192 scores into LDS via the Tensor Data Mover ----
#if defined(__HIP_DEVICE_COMPILE__) && __has_builtin(__builtin_amdgcn_tensor_load_to_lds)
    if (wid == 0) {
        unsigned lds_off = (unsigned)(unsigned long long)(uintptr_t)(&s_lds[0]);
        unsigned long long ga = (unsigned long long)(uintptr_t)(scores + (size_t)b * NN);
        u32x4 g0;
        g0[0] = 1u;                                        // count=1, user desc
        g0[1] = lds_off;                                   // lds_addr
        g0[2] = (unsigned)(ga & 0xffffffffull);            // global_addr[31:0]
        g0[3] = (unsigned)((ga >> 32) & 0x1ffffffull)      // global_addr[56:32]
              | (2u << 30);                                // type = image(2)
        i32x8 g1;
        g1[0] = (int)(2u << 16);                           // data_size = 4B, wg_mask=0
        g1[1] = (int)((NN & 0xffffu) << 16);               // tensor_dim0[15:0]
        g1[2] = (int)((NN >> 16) | (1u << 16));            // tensor_dim0[31:16], tensor_dim1=1
        g1[3] = (int)((NN & 0xffffu) << 16);               // tile_dim0 = 8192
        g1[4] = 0; g1[5] = 0; g1[6] = 0; g1[7] = 0;        // tile_dim1/2=0, strides=0
        i32x4 z4; z4[0] = 0; z4[1] = 0; z4[2] = 0; z4[3] = 0;
        i32x8 z8; z8[0] = 0; z8[1] = 0; z8[2] = 0; z8[3] = 0;
        z8[4] = 0; z8[5] = 0; z8[6] = 0; z8[7] = 0;
        __builtin_amdgcn_tensor_load_to_lds(g0, g1, z4, z4, z8, 0);
    }
#else
    for (int i = tid; i < NN; i += 1024) s_lds[i] = scores[(size_t)b * NN + i];
#endif

    // ---- global scalars -> cost normalization ----
    const float mx   = ws[0];
    const float mn   = ws[1];
    const float ng   = ws[2];
    const float fill = mn - (mx - mn);
    const float smin = (ng > 0.0f) ? fminf(mn, fill) : mn;
    const float cmax = fmaxf(fmaxf(smin * smin, mx * mx),
                             fmaxf((smin - 16.0f) * (smin - 16.0f),
                                   (mx   - 16.0f) * (mx   - 16.0f)));
    const float inv_c = 1.0f / (EPSI * cmax);
    const float MU    = 1.0f / (float)NN;
    const float NU    = 1.0f / (float)NN;
    const float NU16  = (float)(NN - KK) / (float)NN;

    // ---- w^(0)_j = v0_j * B_j = exp(-a_j^2/c)/17,  a_j = 16-j (per-thread regs) ----
    float w[KP1];
    #pragma unroll
    for (int j = 0; j < KP1; ++j) {
        float a = (float)(KK - j);
        w[j] = __expf(-a * a * inv_c) * (1.0f / (float)KP1);
    }

#if defined(__HIP_DEVICE_COMPILE__) && __has_builtin(__builtin_amdgcn_s_wait_tensorcnt)
    __builtin_amdgcn_s_wait_tensorcnt(0);
#endif
    __syncthreads();

    // ---- per-row t_i = exp(2 s_i / c) ----
    float treg[ROWS], preg[ROWS];
    #pragma unroll
    for (int r = 0; r < ROWS; ++r) {
        float sv = s_lds[tid + 1024 * r];
        sv = (sv == -__builtin_inff()) ? fill : sv;
        treg[r] = __expf(2.0f * sv * inv_c);
        preg[r] = 1.0f;
    }

    const int col = lane & 15;
    const int hf  = lane >> 4;
    const int j_of_lane  = KK - col;                 // 16..1
    const float nuj_lane = (col == 0) ? NU16 : NU;   // nu_(16-col)

    // ---- 200 Sinkhorn iterations, one workgroup barrier each ----
    for (int it = 0; it < ITERS; ++it) {
        float* redp = &red[it & 1][0];

        float acc[KP1];
        #pragma unroll
        for (int m = 0; m < KP1; ++m) acc[m] = 0.0f;

        #pragma unroll
        for (int r = 0; r < ROWS; ++r) {
            float t = treg[r];
            float p = w[0];
            #pragma unroll
            for (int j = 1; j < KP1; ++j) p = __builtin_fmaf(p, t, w[j]);
            preg[r] = p;
            float g  = __builtin_amdgcn_rcpf(p) * MU;   // mu / p_i
            float pw = g;
            acc[0] += g;
            #pragma unroll
            for (int m = 1; m < KP1; ++m) { pw *= t; acc[m] += pw; }
        }

        // intra-wave reduction: pure-VALU DPP butterflies (no LDS traffic)
        #pragma unroll
        for (int m = 0; m < KP1; ++m) acc[m] = wave_sum(acc[m]);

        if (lane == 0) {
            #pragma unroll
            for (int m = 0; m < KP1; ++m) redp[wid * RSTR + m] = acc[m];
        }
        __syncthreads();

        // every wave: 32x17 cross-wave reduce via 8 chained WMMAs (ones x B)
        v2f av; av.x = 1.0f; av.y = 1.0f;
        v8f c = {};
        #pragma unroll
        for (int s8 = 0; s8 < 8; ++s8) {
            int k0 = s8 * 4 + hf * 2;                  // 4 partial rows per WMMA
            v2f bv;
            bv.x = redp[(k0 + 0) * RSTR + col];
            bv.y = redp[(k0 + 1) * RSTR + col];
            c = __builtin_amdgcn_wmma_f32_16x16x4_f32(
                    false, av, false, bv, (short)0, c, false, false);
        }
        float csum = c[0];                             // M_col in every lane

        float m16 = wave_sum(redp[lane * RSTR + 16]);  // moment 16

        // new w: lane holds w_(16-col); broadcast via v_readlane into all lanes
        float wl = nuj_lane / csum;
        float w0 = NU / m16;
        w[0] = w0;
        #pragma unroll
        for (int j = 1; j < KP1; ++j) w[j] = bcast_lane(wl, KK - j);
    }

    // ---- emit Gamma[:, :, :16] * n = t^(16-j) * w_j / p_i ----
    const size_t base = (size_t)b * NN * KK;
    #pragma unroll
    for (int r = 0; r < ROWS; ++r) {
        const int i   = tid + 1024 * r;
        const float t = treg[r];
        const float inv = __builtin_amdgcn_rcpf(preg[r]);
        float row[KK];
        float pw = inv;
        #pragma unroll
        for (int m = 1; m <= KK; ++m) { pw *= t; row[KK - m] = pw * w[KK - m]; }
        f4* dst = (f4*)(out + base + (size_t)i * KK);
        #pragma unroll
        for (int q = 0; q < 4; ++q) {
            f4 v; v.x = row[4 * q]; v.y = row[4 * q + 1];
            v.z = row[4 * q + 2];   v.w = row[4 * q + 3];
            __builtin_nontemporal_store(v, dst + q);
        }
    }
}

// ---------------------------------------------------------------------------
extern "C" void kernel_launch(void* const* d_in, const int* in_sizes, int n_in,
                              void* d_out, int out_size, void* d_ws, size_t ws_size,
                              hipStream_t stream) {
    (void)n_in; (void)out_size; (void)ws_size;
    const float* scores = (const float*)d_in[0];
    float* out = (float*)d_out;
    float* ws  = (float*)d_ws;
    const int ntot = in_sizes[0];     // BS * NN

    minmax_part <<<256, 256, 0, stream>>>(scores, ws, ntot);
    minmax_final<<<1,   256, 0, stream>>>(ws);
    sinkhorn_topk<<<BS, 1024, 0, stream>>>(scores, out, ws);
}